// TorchWelfordEstimator_64673617543204
// MI455X (gfx1250) — compile-verified
//
#include <hip/hip_runtime.h>
#include <stdint.h>

// Welford batch scan for MI455X (gfx1250).
// Memory-bound streaming kernel: ~0.52 GiB traffic -> ~23us floor @ 23.3 TB/s.
// Uses CDNA5 async global->LDS loads (ASYNCcnt) as an 8-deep per-wave ring to
// keep HBM saturated; no cross-wave sharing so no barriers are needed.

typedef float v4f __attribute__((ext_vector_type(4)));
typedef int   v4i __attribute__((ext_vector_type(4)));

#define BLOCK 256
#define VEC 4
#define NPB (BLOCK * VEC)          // neurons per block = 1024
#define DEPTH 8                    // async pipeline stages
#define STAGE_FLOATS (BLOCK * VEC) // 1024 floats = 4KB per stage
#define STAGE_BYTES (STAGE_FLOATS * 4)

// CDNA5 async copy: per-lane 16B global -> LDS, tracked by ASYNCcnt.
// GV addressing mode: 64-bit vaddr pair, saddr = off.
__device__ __forceinline__ void async_ld_b128(unsigned lds_byte_off, const float* g) {
  asm volatile("global_load_async_to_lds_b128 %0, %1, off"
               :: "v"(lds_byte_off),
                  "v"((unsigned long long)(uintptr_t)g)
               : "memory");
}

__global__ __launch_bounds__(BLOCK) void welford_scan_kernel(
    const float* __restrict__ x,
    const float* __restrict__ m0,
    const float* __restrict__ s0,
    const int*   __restrict__ nn0,
    const int*   __restrict__ nsamp,
    float* __restrict__ out_x,
    float* __restrict__ out_m,
    float* __restrict__ out_s,
    int*   __restrict__ out_nn,
    int*   __restrict__ out_n,
    int N, int B)
{
  __shared__ float smem[DEPTH * STAGE_FLOATS]; // 32 KB ring buffer

  const int tid   = threadIdx.x;
  const int gbase = blockIdx.x * NPB;
  const int gid   = gbase + tid * VEC;
  const int n0    = nsamp[0];

  if (blockIdx.x == 0 && tid == 0) out_n[0] = n0 + B;

  if (gbase + NPB <= N) {
    // ---- fast path: full block, async-pipelined ----
    v4f m  = *(const v4f*)(m0  + gid);
    v4f s  = *(const v4f*)(s0  + gid);
    v4i nn = *(const v4i*)(nn0 + gid);

    // LDS byte offset for this lane (low 32 bits of flat addr == LDS offset).
    const unsigned lds_lane = ((unsigned)(uintptr_t)smem) + (unsigned)tid * 16u;

    // Prime the pipeline: samples 0..DEPTH-1.
#pragma unroll
    for (int d = 0; d < DEPTH; ++d)
      if (d < B)
        async_ld_b128(lds_lane + (unsigned)d * STAGE_BYTES,
                      x + (size_t)d * N + gid);

#define PROCESS(bb)                                                            \
    {                                                                          \
      const int stage = (bb) & (DEPTH - 1);                                    \
      v4f v = ((const v4f*)smem)[stage * BLOCK + tid];                         \
      /* passthrough copy of x, streaming (non-temporal) */                    \
      __builtin_nontemporal_store(v, (v4f*)(out_x + (size_t)(bb) * N + gid));  \
      const float inv = 1.0f / (float)(n0 + (bb) + 1);                         \
      _Pragma("unroll")                                                        \
      for (int j = 0; j < VEC; ++j) {                                          \
        nn[j] += (v[j] != 0.0f) ? 1 : 0;                                       \
        const float old = m[j];                                                \
        const float mn  = old + (v[j] - old) * inv;                            \
        m[j] = mn;                                                             \
        s[j] += (v[j] - mn) * (v[j] - old);                                    \
      }                                                                        \
    }

    int b = 0;
    const int main_end = (B > DEPTH) ? (B - DEPTH) : 0;
    for (; b < main_end; ++b) {
      // Async loads complete in order: <=DEPTH-1 outstanding means the load
      // for sample b (issued DEPTH iterations ago) has landed in LDS.
      asm volatile("s_wait_asynccnt %0" :: "i"(DEPTH - 1) : "memory");
      PROCESS(b);
      // Refill the stage just consumed with sample b+DEPTH.
      async_ld_b128(lds_lane + (unsigned)(b & (DEPTH - 1)) * STAGE_BYTES,
                    x + (size_t)(b + DEPTH) * N + gid);
    }
    // Drain: everything issued, wait for all, then consume the tail stages.
    asm volatile("s_wait_asynccnt 0" ::: "memory");
    for (; b < B; ++b) {
      PROCESS(b);
    }
#undef PROCESS

    *(v4f*)(out_m  + gid) = m;
    *(v4f*)(out_s  + gid) = s;
    *(v4i*)(out_nn + gid) = nn;
  } else {
    // ---- scalar tail path (not hit for N = 64^3) ----
    for (int j = 0; j < VEC; ++j) {
      const int idx = gid + j;
      if (idx >= N) break;
      float mm = m0[idx], ss = s0[idx];
      int c = nn0[idx];
      for (int b = 0; b < B; ++b) {
        const float v = x[(size_t)b * N + idx];
        out_x[(size_t)b * N + idx] = v;
        c += (v != 0.0f) ? 1 : 0;
        const float old = mm;
        mm = old + (v - old) / (float)(n0 + b + 1);
        ss += (v - mm) * (v - old);
      }
      out_m[idx] = mm; out_s[idx] = ss; out_nn[idx] = c;
    }
  }
}

extern "C" void kernel_launch(void* const* d_in, const int* in_sizes, int n_in,
                              void* d_out, int out_size, void* d_ws, size_t ws_size,
                              hipStream_t stream) {
  (void)n_in; (void)out_size; (void)d_ws; (void)ws_size;

  const float* x   = (const float*)d_in[0]; // (B, C, H, W) f32
  const float* m0  = (const float*)d_in[1]; // (C, H, W) f32
  const float* s0  = (const float*)d_in[2]; // (C, H, W) f32
  const int*   nn0 = (const int*)  d_in[3]; // (C, H, W) i32
  const int*   nsm = (const int*)  d_in[4]; // (1,) i32

  const int N = in_sizes[1];                    // C*H*W
  const int B = (N > 0) ? in_sizes[0] / N : 0;  // batch

  // Outputs concatenated flat in return order: x, m, s, nn, n.
  float* out_x  = (float*)d_out;
  float* out_m  = out_x + (size_t)B * N;
  float* out_s  = out_m + N;
  int*   out_nn = (int*)(out_s + N);
  int*   out_n  = out_nn + N;

  if (N <= 0 || B <= 0) return;

  const int grid = (N + NPB - 1) / NPB;
  welford_scan_kernel<<<dim3(grid), dim3(BLOCK), 0, stream>>>(
      x, m0, s0, nn0, nsm, out_x, out_m, out_s, out_nn, out_n, N, B);
}